// CAGAT_MinSum_Layer_True_88802743812477
// MI455X (gfx1250) — compile-verified
//
#include <hip/hip_runtime.h>
#include <hip/hip_bf16.h>
#include <stdint.h>

#define N_NODES 50000
#define N_EDGES 1600000
#define HEADS 8
#define EPSV 1e-16f

typedef float v2f __attribute__((ext_vector_type(2)));
typedef float v8f __attribute__((ext_vector_type(8)));

// Monotone bijection float -> uint32 so unsigned atomicMax == float max.
__device__ __forceinline__ unsigned fenc(float f) {
    unsigned u = __float_as_uint(f);
    return u ^ ((unsigned)((int)u >> 31) | 0x80000000u);
}
__device__ __forceinline__ float fdec(unsigned u) {
    unsigned b = (u & 0x80000000u) ? (u ^ 0x80000000u) : ~u;
    return __uint_as_float(b);
}

// ---------------------------------------------------------------------------
// Coefficient precompute via V_WMMA_F32_16X16X4_F32 (single wave32).
//
// A-matrix (16x64, consumed 16x4 per WMMA): rows 0..7  = W_att[h, 0:64]
//                                           rows 8..15 = W_att[h, 64:128]
// B-matrix (64x2 padded to 64x16):          col 0 = w = W_proj[:,0]
//                                           col 1 = b_proj
// Result D (16x16): D[h,0]=A[h], D[h+8,0]=B[h],
//                   D[h,1]+D[h+8,1]+b_att[h] = Dcoef[h]
//
// VGPR stripings per CDNA5 ISA:
//   A 16x4 f32 : lane L<16: {K=k0,k0+1}; lane L>=16: {K=k0+2,k0+3}, M=L&15
//   B 4x16 f32 : lanes 0-15 hold K=k0,k0+1 ; lanes 16-31 hold K=k0+2,k0+3, N=L&15
//   D 16x16    : VGPR r, lanes 0-15 -> (M=r, N=lane); lanes 16-31 -> (M=8+r, N=lane-16)
//
// coef layout (floats): [0:8)=A  [8:16)=B  [16:24)=C  [24:32)=D  [32:40)=penalty
// ---------------------------------------------------------------------------
__global__ void precompute_coefs(const float* __restrict__ Watt,   // (8,129)
                                 const float* __restrict__ w,      // (64)
                                 const float* __restrict__ bproj,  // (64)
                                 const float* __restrict__ batt,   // (8)
                                 const float* __restrict__ pen,    // (8)
                                 float* __restrict__ coef)
{
    __shared__ float stage[16][16];
    const int lane = threadIdx.x;          // 0..31
    const int mn   = lane & 15;            // A-row / B-col index for this lane
    const int half = lane >> 4;            // 0 or 1 (K sub-block)
    const float* row = (mn < 8) ? (Watt + mn * 129)
                                : (Watt + (mn - 8) * 129 + 64);
    v8f c = {0.f, 0.f, 0.f, 0.f, 0.f, 0.f, 0.f, 0.f};
#pragma unroll
    for (int t = 0; t < 16; ++t) {
        const int kk = t * 4 + half * 2;
        v2f a;
        a.x = row[kk];
        a.y = row[kk + 1];
        v2f b;
        b.x = (mn == 0) ? w[kk]     : ((mn == 1) ? bproj[kk]     : 0.f);
        b.y = (mn == 0) ? w[kk + 1] : ((mn == 1) ? bproj[kk + 1] : 0.f);
        c = __builtin_amdgcn_wmma_f32_16x16x4_f32(false, a, false, b,
                                                  (short)0, c, false, false);
    }
#pragma unroll
    for (int r = 0; r < 8; ++r)
        stage[r + half * 8][mn] = c[r];
    __syncthreads();
    if (lane < 8) {
        const int h = lane;
        coef[h]      = stage[h][0];                              // A
        coef[8 + h]  = stage[h + 8][0];                          // B
        coef[16 + h] = Watt[h * 129 + 128];                      // C
        coef[24 + h] = stage[h][1] + stage[h + 8][1] + batt[h];  // D
        coef[32 + h] = pen[h];                                   // penalty
    }
}

// ---------------------------------------------------------------------------
__global__ void init_state(unsigned* __restrict__ segmax,
                           float* __restrict__ segsum,
                           float* __restrict__ out, int n8, int n)
{
    const int i = blockIdx.x * blockDim.x + threadIdx.x;
    if (i < n8) { segmax[i] = 0u; segsum[i] = 0.f; }  // 0 < fenc(any float)
    if (i < n)  out[i] = 0.f;
}

__device__ __forceinline__ float edge_raw(float fs, float fd, float cm, int h,
                                          const float* __restrict__ coef)
{
    float raw = fmaf(fs, coef[h], fmaf(fd, coef[8 + h],
                 fmaf(cm, coef[16 + h], coef[24 + h])));
    raw = (raw >= 0.f) ? raw : 0.2f * raw;           // leaky relu
    return fmaf(cm, coef[32 + h], raw);              // + cycle penalty
}

// pass 1: per-(edge,head) segment max via u32-encoded atomicMax (L2-resident)
__global__ void pass_max(const long long* __restrict__ ei,
                         const float* __restrict__ nf,
                         const float* __restrict__ cyc,
                         const float* __restrict__ coef,
                         unsigned* __restrict__ segmax)
{
    const int tid = blockIdx.x * blockDim.x + threadIdx.x;
    if (tid >= N_EDGES * HEADS) return;
    const int e = tid >> 3;
    const int h = tid & 7;
    const long long s = ei[e];
    const long long d = ei[N_EDGES + e];
    const float cm = cyc[e];
    const float raw = edge_raw(nf[s], nf[d], cm, h, coef);
    atomicMax(segmax + (int)d * HEADS + h, fenc(raw));
}

// pass 2: per-(edge,head) exp-sum
__global__ void pass_sum(const long long* __restrict__ ei,
                         const float* __restrict__ nf,
                         const float* __restrict__ cyc,
                         const float* __restrict__ coef,
                         const unsigned* __restrict__ segmax,
                         float* __restrict__ segsum)
{
    const int tid = blockIdx.x * blockDim.x + threadIdx.x;
    if (tid >= N_EDGES * HEADS) return;
    const int e = tid >> 3;
    const int h = tid & 7;
    const long long s = ei[e];
    const long long d = ei[N_EDGES + e];
    const float cm = cyc[e];
    const float raw = edge_raw(nf[s], nf[d], cm, h, coef);
    const int idx = (int)d * HEADS + h;
    atomicAdd(segsum + idx, __expf(raw - fdec(segmax[idx])));
}

// pass 3: per-edge head-mean of attention, scaled scatter-add into out
__global__ void pass_agg(const long long* __restrict__ ei,
                         const float* __restrict__ nf,
                         const float* __restrict__ cyc,
                         const float* __restrict__ coef,
                         const unsigned* __restrict__ segmax,
                         const float* __restrict__ segsum,
                         const float* __restrict__ scaler,
                         float* __restrict__ out)
{
    const int e = blockIdx.x * blockDim.x + threadIdx.x;
    if (e >= N_EDGES) return;
    const long long s = ei[e];
    const long long d = ei[N_EDGES + e];
    const float fs = nf[s];
    const float fd = nf[d];
    const float cm = cyc[e];
    const int base = (int)d * HEADS;
    float acc = 0.f;
#pragma unroll
    for (int h = 0; h < HEADS; ++h) {
        const float raw = edge_raw(fs, fd, cm, h, coef);
        const float ev  = __expf(raw - fdec(segmax[base + h]));
        acc += ev / (segsum[base + h] + EPSV);
    }
    const float msg = fs * (acc * (1.0f / HEADS)) * scaler[0];
    atomicAdd(out + (int)d, msg);
}

// ---------------------------------------------------------------------------
extern "C" void kernel_launch(void* const* d_in, const int* in_sizes, int n_in,
                              void* d_out, int out_size, void* d_ws, size_t ws_size,
                              hipStream_t stream)
{
    (void)in_sizes; (void)n_in; (void)out_size; (void)ws_size;
    const float*     nf   = (const float*)d_in[0];      // node_features (N)
    const float*     cyc  = (const float*)d_in[1];      // cycle_mask (E)
    const float*     Wp   = (const float*)d_in[2];      // W_proj (64,1) -> w
    const float*     bp   = (const float*)d_in[3];      // b_proj (64)
    const float*     Wa   = (const float*)d_in[4];      // W_att (8,129)
    const float*     ba   = (const float*)d_in[5];      // b_att (8)
    const float*     pen  = (const float*)d_in[6];      // cycle_penalty (8)
    const float*     scal = (const float*)d_in[7];      // min_sum_scaler (1)
    const long long* ei   = (const long long*)d_in[8];  // edge_index (2,E) int64
    float* out = (float*)d_out;

    char* ws = (char*)d_ws;
    float*    coef   = (float*)ws;                                   // 40 floats (pad 256B)
    unsigned* segmax = (unsigned*)(ws + 256);                        // N*8 u32
    float*    segsum = (float*)(ws + 256 + (size_t)N_NODES * HEADS * 4);

    precompute_coefs<<<1, 32, 0, stream>>>(Wa, Wp, bp, ba, pen, coef);

    const int n8 = N_NODES * HEADS;
    init_state<<<(n8 + 255) / 256, 256, 0, stream>>>(segmax, segsum, out, n8, N_NODES);

    const int eh = N_EDGES * HEADS;
    pass_max<<<(eh + 255) / 256, 256, 0, stream>>>(ei, nf, cyc, coef, segmax);
    pass_sum<<<(eh + 255) / 256, 256, 0, stream>>>(ei, nf, cyc, coef, segmax, segsum);
    pass_agg<<<(N_EDGES + 255) / 256, 256, 0, stream>>>(ei, nf, cyc, coef,
                                                        segmax, segsum, scal, out);
}